// PointNet_SA_Module1_54099408060832
// MI455X (gfx1250) — compile-verified
//
#include <hip/hip_runtime.h>
#include <hip/hip_bf16.h>
#include <math.h>

#define BATCH 8
#define NPTS  2048
#define KNN   20
#define EPSC  1e-5f

typedef __attribute__((ext_vector_type(2))) float v2f;
typedef __attribute__((ext_vector_type(8))) float v8f;

__device__ __forceinline__ v8f wmma4(v2f a, v2f b, v8f c) {
  // D = A(16x4) * B(4x16) + C(16x16), fp32, wave32
  return __builtin_amdgcn_wmma_f32_16x16x4_f32(false, a, false, b, (short)0, c,
                                               false, false);
}

__device__ __forceinline__ float lrelu(float x) { return x > 0.f ? x : 0.2f * x; }
__device__ __forceinline__ float siluf(float x) { return x / (1.f + expf(-x)); }

// ---------------------------------------------------------------- transpose
// x (B,3,N) -> xc (B,N,3)
__global__ void k_transpose(const float* __restrict__ x, float* __restrict__ xc) {
  int i = blockIdx.x * blockDim.x + threadIdx.x;
  if (i >= BATCH * 3 * NPTS) return;
  int b = i / (NPTS * 3), r = i % (NPTS * 3), n = r / 3, c = r % 3;
  xc[i] = x[((size_t)b * 3 + c) * NPTS + n];
}

// ---------------------------------------------------------------- build f
// stage feature: f[b,n,:] = [xc | p_prev], padded to CP (4 or 68)
__global__ void k_build_f(const float* __restrict__ xc, const float* __restrict__ pprev,
                          float* __restrict__ f, int CP) {
  int i = blockIdx.x * blockDim.x + threadIdx.x;
  if (i >= BATCH * NPTS) return;
  float* fr = f + (size_t)i * CP;
  fr[0] = xc[i * 3 + 0];
  fr[1] = xc[i * 3 + 1];
  fr[2] = xc[i * 3 + 2];
  if (CP == 4) {
    fr[3] = 0.f;
  } else {
    for (int j = 0; j < 64; ++j) fr[3 + j] = pprev[(size_t)i * 64 + j];
    fr[67] = 0.f;
  }
}

// ---------------------------------------------------------------- fused knn
// One wave per 16-row tile: WMMA Gram -> dist -> streaming top-20 in LDS.
__global__ void k_knn(const float* __restrict__ f, int CP, int* __restrict__ idxo) {
  __shared__ float As[16][68];
  __shared__ float Bs[16][68];
  __shared__ float sqA[16];
  __shared__ float sqB[16];
  __shared__ float dtile[16][16];
  __shared__ float topd[16][KNN];
  __shared__ int   topi[16][KNN];

  int b  = blockIdx.x >> 7;       // NPTS/16 = 128 row tiles per batch
  int rt = blockIdx.x & 127;
  int r0 = rt * 16;
  int lane = threadIdx.x;
  const float* fb = f + (size_t)b * NPTS * CP;

  for (int i = lane; i < 16 * CP; i += 32) {
    int r = i / CP, c = i % CP;
    As[r][c] = fb[(size_t)(r0 + r) * CP + c];
  }
  __syncthreads();
  if (lane < 16) {
    float s = 0.f;
    for (int c = 0; c < CP; ++c) s += As[lane][c] * As[lane][c];
    sqA[lane] = s;
    for (int j = 0; j < KNN; ++j) { topd[lane][j] = -3e30f; topi[lane][j] = 0; }
  }
  __syncthreads();

  int m = lane & 15, kh = (lane >> 4) * 2, half = lane >> 4;
  for (int ct = 0; ct < NPTS / 16; ++ct) {
    int c0 = ct * 16;
    for (int i = lane; i < 16 * CP; i += 32) {
      int r = i / CP, c = i % CP;
      Bs[r][c] = fb[(size_t)(c0 + r) * CP + c];
    }
    __syncthreads();
    if (lane < 16) {
      float s = 0.f;
      for (int c = 0; c < CP; ++c) s += Bs[lane][c] * Bs[lane][c];
      sqB[lane] = s;
    }
    __syncthreads();

    v8f acc = {};
    for (int k = 0; k < CP; k += 4) {
      v2f a;  a.x  = As[m][k + kh]; a.y  = As[m][k + kh + 1];
      v2f bb; bb.x = Bs[m][k + kh]; bb.y = Bs[m][k + kh + 1];
      acc = wmma4(a, bb, acc);
    }
#pragma unroll
    for (int v = 0; v < 8; ++v) {
      int M = v + half * 8;
      dtile[M][m] = 2.f * acc[v] - sqA[M] - sqB[m];   // pd (larger = nearer)
    }
    __syncthreads();
    if (lane < 16) {
      for (int j = 0; j < 16; ++j) {
        float d = dtile[lane][j];
        if (d > topd[lane][KNN - 1]) {
          int pos = KNN - 1;
          while (pos > 0 && topd[lane][pos - 1] < d) {
            topd[lane][pos] = topd[lane][pos - 1];
            topi[lane][pos] = topi[lane][pos - 1];
            --pos;
          }
          topd[lane][pos] = d;
          topi[lane][pos] = c0 + j;
        }
      }
    }
    __syncthreads();
  }
  for (int i = lane; i < 16 * KNN; i += 32) {
    int r = i / KNN, j = i % KNN;
    idxo[((size_t)b * NPTS + r0 + r) * KNN + j] = topi[r][j];
  }
}

// ---------------------------------------------------------------- edge conv
// One wave per point: e(20x2C) @ W(2Cx64) via WMMA, BN+lrelu, max over k.
__global__ void k_edgeconv(const float* __restrict__ f, int C, int CP2, int FS,
                           const int* __restrict__ idx, const float* __restrict__ Wt,
                           const float* __restrict__ g, const float* __restrict__ bv,
                           float* __restrict__ pout) {
  __shared__ float e[32][136];
  int b = blockIdx.x >> 11;
  int n = blockIdx.x & (NPTS - 1);
  int lane = threadIdx.x;
  const float* fb  = f + (size_t)b * NPTS * FS;
  const float* ctr = fb + (size_t)n * FS;
  const int*   id  = idx + ((size_t)b * NPTS + n) * KNN;

  for (int i = lane; i < 32 * CP2; i += 32) {
    int r = i / CP2, c = i % CP2;
    float val = 0.f;
    if (r < KNN) {
      if (c < C) {
        int nb = id[r];
        val = fb[(size_t)nb * FS + c] - ctr[c];
      } else if (c < 2 * C) {
        val = ctr[c - C];
      }
    }
    e[r][c] = val;
  }
  __syncthreads();

  int m = lane & 15, kh = (lane >> 4) * 2, half = lane >> 4;
  int C2 = 2 * C;
  for (int nt = 0; nt < 4; ++nt) {
    int o = nt * 16 + m;
    v8f acc0 = {}, acc1 = {};
    for (int k = 0; k < CP2; k += 4) {
      int kb = k + kh;
      v2f bb;
      bb.x = (kb     < C2) ? Wt[(size_t)kb * 64 + o]       : 0.f;
      bb.y = (kb + 1 < C2) ? Wt[(size_t)(kb + 1) * 64 + o] : 0.f;
      v2f a0; a0.x = e[m][kb];      a0.y = e[m][kb + 1];
      v2f a1; a1.x = e[16 + m][kb]; a1.y = e[16 + m][kb + 1];
      acc0 = wmma4(a0, bb, acc0);
      acc1 = wmma4(a1, bb, acc1);
    }
    float gs = g[o] * rsqrtf(1.f + EPSC), bo = bv[o];
    float mx = -3e30f;
#pragma unroll
    for (int v = 0; v < 8; ++v) {
      mx = fmaxf(mx, lrelu(acc0[v] * gs + bo));           // rows 0..15 all valid
      int Mr = 16 + v + half * 8;                          // rows 16..31
      if (Mr < KNN) mx = fmaxf(mx, lrelu(acc1[v] * gs + bo));
    }
    float oth = __shfl_xor(mx, 16, 32);                    // combine M-halves
    mx = fmaxf(mx, oth);
    if (lane < 16) pout[((size_t)b * NPTS + n) * 64 + o] = mx;
  }
}

// ---------------------------------------------------------------- pooling
__global__ void k_pool(const float* __restrict__ P, float* __restrict__ mx,
                       float* __restrict__ mn) {
  int b = blockIdx.x >> 2, s = blockIdx.x & 3, o = threadIdx.x;
  const float* p = P + ((size_t)s * BATCH * NPTS + (size_t)b * NPTS) * 64;
  float m = -3e30f, sum = 0.f;
  for (int n = 0; n < NPTS; ++n) {
    float v = p[(size_t)n * 64 + o];
    m = fmaxf(m, v);
    sum += v;
  }
  mx[(b * 4 + s) * 64 + o] = m;
  mn[(b * 4 + s) * 64 + o] = sum / (float)NPTS;
}

// ---------------------------------------------------------------- mamba2 (L=4, tiny)
__global__ void k_mamba(const float* __restrict__ inp, const float* __restrict__ inW,
                        const float* __restrict__ convw, const float* __restrict__ convb,
                        const float* __restrict__ dtb, const float* __restrict__ Alog,
                        const float* __restrict__ Dp, const float* __restrict__ normw,
                        const float* __restrict__ outW, float* __restrict__ yout) {
  __shared__ float u[4][64];
  __shared__ float zx[4][514];
  __shared__ float xbc[4][384];
  __shared__ float dts[4][2];
  __shared__ float ybuf[4][128];
  __shared__ float rmsv[4];
  int b = blockIdx.x, t = threadIdx.x;

  for (int i = t; i < 4 * 64; i += 128) u[i / 64][i % 64] = inp[b * 256 + i];
  __syncthreads();
  for (int i = t; i < 4 * 514; i += 128) {
    int l = i / 514, j = i % 514;
    float s = 0.f;
    for (int c = 0; c < 64; ++c) s += u[l][c] * inW[c * 514 + j];
    zx[l][j] = s;
  }
  __syncthreads();
  if (t < 8) {
    int l = t >> 1, h = t & 1;
    float v = zx[l][512 + h] + dtb[h];
    dts[l][h] = (v > 20.f) ? v : log1pf(expf(v));          // softplus
  }
  for (int i = t; i < 4 * 384; i += 128) {
    int l = i / 384, j = i % 384;
    float s = convb[j];
    for (int q = 0; q < 4; ++q) {
      int ll = l + q - 3;
      if (ll >= 0) s += zx[ll][128 + j] * convw[q * 384 + j];
    }
    xbc[l][j] = siluf(s);
  }
  __syncthreads();

  int hd = t >> 6, pp = t & 63;
  float hstate[128];
  for (int nn = 0; nn < 128; ++nn) hstate[nn] = 0.f;
  float nA = -expf(Alog[hd]);
  float yv[4];
  for (int l = 0; l < 4; ++l) {
    float dtl = dts[l][hd];
    float dA  = expf(dtl * nA);
    float xv  = xbc[l][hd * 64 + pp];
    float dx  = dtl * xv;
    float acc = 0.f;
    for (int nn = 0; nn < 128; ++nn) {
      float hh = hstate[nn] * dA + dx * xbc[l][128 + nn];  // * B
      hstate[nn] = hh;
      acc += hh * xbc[l][256 + nn];                        // * C
    }
    yv[l] = acc + Dp[hd] * xv;
  }
  for (int l = 0; l < 4; ++l) {
    float z = zx[l][t];                                    // gate channel = t
    ybuf[l][t] = yv[l] * siluf(z);
  }
  __syncthreads();
  if (t < 4) {
    float s = 0.f;
    for (int ch = 0; ch < 128; ++ch) { float v = ybuf[t][ch]; s += v * v; }
    rmsv[t] = rsqrtf(s / 128.f + EPSC);
  }
  __syncthreads();
  for (int i = t; i < 4 * 64; i += 128) {
    int l = i / 64, o = i % 64;
    float s = 0.f;
    for (int ch = 0; ch < 128; ++ch)
      s += ybuf[l][ch] * rmsv[l] * normw[ch] * outW[ch * 64 + o];
    yout[b * 256 + l * 64 + o] = s;
  }
}

// ---------------------------------------------------------------- fuse -> 4 weights
__global__ void k_fuse(const float* __restrict__ y1, const float* __restrict__ y2,
                       const float* f1W1, const float* f1W2, const float* f1W3,
                       const float* f1b1, const float* f1b2, const float* f1b3,
                       const float* f1g1, const float* f1g2,
                       const float* f2W1, const float* f2W2, const float* f2W3,
                       const float* f2b1, const float* f2b2, const float* f2b3,
                       const float* f2g1, const float* f2g2,
                       float* __restrict__ wv) {
  __shared__ float acc[2][32];
  int t = threadIdx.x;
  if (t < 64) {
    int set = t >> 5, i = t & 31, b = i >> 2, l = i & 3;
    const float* y   = set ? y2   : y1;
    const float* W1p = set ? f2W1 : f1W1;
    const float* W2p = set ? f2W2 : f1W2;
    const float* W3p = set ? f2W3 : f1W3;
    const float* b1p = set ? f2b1 : f1b1;
    const float* b2p = set ? f2b2 : f1b2;
    const float* b3p = set ? f2b3 : f1b3;
    const float* g1p = set ? f2g1 : f1g1;
    const float* g2p = set ? f2g2 : f1g2;
    float gsc = rsqrtf(1.f + EPSC);
    float h1[64];
    for (int o = 0; o < 64; ++o) {
      float s = 0.f;
      for (int c = 0; c < 64; ++c) s += y[b * 256 + l * 64 + c] * W1p[c * 64 + o];
      h1[o] = lrelu(s * g1p[o] * gsc + b1p[o]);
    }
    float h2[8];
    for (int q = 0; q < 8; ++q) {
      float s = 0.f;
      for (int o = 0; o < 64; ++o) s += h1[o] * W2p[o * 8 + q];
      h2[q] = lrelu(s * g2p[q] * gsc + b2p[q]);
    }
    float w = b3p[0];
    for (int q = 0; q < 8; ++q) w += h2[q] * W3p[q];
    acc[set][i] = w;
  }
  __syncthreads();
  if (t < 4) {
    float s = 0.f;
    for (int b = 0; b < BATCH; ++b) s += acc[0][b * 4 + t] + acc[1][b * 4 + t];
    wv[t] = s / (float)BATCH;
  }
}

// ---------------------------------------------------------------- final 256->1024
// One wave per 16(n) x 64(o) output tile; K=256 via WMMA f32 16x16x4.
__global__ void k_final(const float* __restrict__ P, const float* __restrict__ wv,
                        const float* __restrict__ W5, const float* __restrict__ g5,
                        const float* __restrict__ b5, float* __restrict__ out) {
  __shared__ float At[16][256];
  int b = blockIdx.x, n0 = blockIdx.y * 16, gt = blockIdx.z;
  int lane = threadIdx.x;
  for (int i = lane; i < 16 * 256; i += 32) {
    int r = i >> 8, c = i & 255;
    int s = c >> 6, j = c & 63;
    At[r][c] = wv[s] * P[(((size_t)s * BATCH + b) * NPTS + (n0 + r)) * 64 + j];
  }
  __syncthreads();
  int m = lane & 15, kh = (lane >> 4) * 2, half = lane >> 4;
  for (int nt = 0; nt < 4; ++nt) {
    int o = gt * 64 + nt * 16 + m;
    __builtin_prefetch(&W5[o], 0, 1);
    v8f acc = {};
    for (int k = 0; k < 256; k += 4) {
      v2f a;  a.x  = At[m][k + kh];            a.y  = At[m][k + kh + 1];
      v2f bb; bb.x = W5[(size_t)(k + kh) * 1024 + o];
      bb.y = W5[(size_t)(k + kh + 1) * 1024 + o];
      acc = wmma4(a, bb, acc);
    }
    float gs = g5[o] * rsqrtf(1.f + EPSC), bo = b5[o];
#pragma unroll
    for (int v = 0; v < 8; ++v) {
      int n = n0 + v + half * 8;
      out[((size_t)b * 1024 + o) * NPTS + n] = lrelu(acc[v] * gs + bo);
    }
  }
}

// ================================================================ launcher
extern "C" void kernel_launch(void* const* d_in, const int* in_sizes, int n_in,
                              void* d_out, int out_size, void* d_ws, size_t ws_size,
                              hipStream_t stream) {
  const int B = BATCH, N = NPTS;
  // x has 8*3*2048 = 49152 elements; params leaves are pytree-sorted.
  int xi = (in_sizes[0] == B * 3 * N) ? 0 : n_in - 1;
  int p0 = (xi == 0) ? 1 : 0;
  const float* x = (const float*)d_in[xi];
  auto PF = [&](int k) { return (const float*)d_in[p0 + k]; };
  const float *W1 = PF(0), *W2 = PF(1), *W3 = PF(2), *W4 = PF(3), *W5 = PF(4);
  const float *b1 = PF(5), *b2 = PF(6), *b3 = PF(7), *b4 = PF(8), *b5 = PF(9);
  const float *f1W1 = PF(10), *f1W2 = PF(11), *f1W3 = PF(12), *f1b1 = PF(13),
              *f1b2 = PF(14), *f1b3 = PF(15), *f1g1 = PF(16), *f1g2 = PF(17);
  const float *f2W1 = PF(18), *f2W2 = PF(19), *f2W3 = PF(20), *f2b1 = PF(21),
              *f2b2 = PF(22), *f2b3 = PF(23), *f2g1 = PF(24), *f2g2 = PF(25);
  const float *g1 = PF(26), *g2 = PF(27), *g3 = PF(28), *g4 = PF(29), *g5 = PF(30);
  const float *m1A = PF(31), *m1D = PF(32), *m1cb = PF(33), *m1cw = PF(34),
              *m1dt = PF(35), *m1in = PF(36), *m1nw = PF(37), *m1out = PF(38);
  const float *m2A = PF(39), *m2D = PF(40), *m2cb = PF(41), *m2cw = PF(42),
              *m2dt = PF(43), *m2in = PF(44), *m2nw = PF(45), *m2out = PF(46);

  const float* Wst[4] = {W1, W2, W3, W4};
  const float* gst[4] = {g1, g2, g3, g4};
  const float* bst[4] = {b1, b2, b3, b4};

  // workspace layout (floats)
  float* ws = (float*)d_ws;
  float* xc   = ws;                                         // B*N*3      = 49152
  float* fbuf = xc + (size_t)B * N * 3;                     // B*N*68     = 1114112
  float* Pbuf = fbuf + (size_t)B * N * 68;                  // 4*B*N*64   = 4194304
  int*   idxb = (int*)(Pbuf + (size_t)4 * B * N * 64);      // B*N*20 ints
  float* mx   = (float*)(idxb + (size_t)B * N * KNN);       // B*4*64
  float* mn   = mx + B * 4 * 64;
  float* y1   = mn + B * 4 * 64;
  float* y2   = y1 + B * 4 * 64;
  float* wv   = y2 + B * 4 * 64;                            // 4 floats

  k_transpose<<<(B * 3 * N + 255) / 256, 256, 0, stream>>>(x, xc);

  for (int s = 0; s < 4; ++s) {
    int CP = (s == 0) ? 4 : 68;      // padded feature stride
    int C  = (s == 0) ? 3 : 67;      // real channels
    int CP2 = (s == 0) ? 8 : 136;    // padded 2C
    const float* pprev = (s == 0) ? xc : (Pbuf + (size_t)(s - 1) * B * N * 64);
    k_build_f<<<(B * N + 255) / 256, 256, 0, stream>>>(xc, pprev, fbuf, CP);
    k_knn<<<B * N / 16, 32, 0, stream>>>(fbuf, CP, idxb);
    k_edgeconv<<<B * N, 32, 0, stream>>>(fbuf, C, CP2, CP, idxb,
                                         Wst[s], gst[s], bst[s],
                                         Pbuf + (size_t)s * B * N * 64);
  }

  k_pool<<<B * 4, 64, 0, stream>>>(Pbuf, mx, mn);
  k_mamba<<<B, 128, 0, stream>>>(mx, m1in, m1cw, m1cb, m1dt, m1A, m1D, m1nw, m1out, y1);
  k_mamba<<<B, 128, 0, stream>>>(mn, m2in, m2cw, m2cb, m2dt, m2A, m2D, m2nw, m2out, y2);
  k_fuse<<<1, 64, 0, stream>>>(y1, y2,
                               f1W1, f1W2, f1W3, f1b1, f1b2, f1b3, f1g1, f1g2,
                               f2W1, f2W2, f2W3, f2b1, f2b2, f2b3, f2g1, f2g2, wv);
  k_final<<<dim3(B, N / 16, 16), 32, 0, stream>>>(Pbuf, wv, W5, g5, b5, (float*)d_out);
}